// GCN_48661979464167
// MI455X (gfx1250) — compile-verified
//
#include <hip/hip_runtime.h>
#include <math.h>

#define DH 128
#define LN_EPS 1e-5f
#define A_STRIDE 132   // 16x128 A-tile row stride in floats; 132%64=4 -> conflict-free b64 LDS reads

typedef __attribute__((ext_vector_type(2))) float v2f;
typedef __attribute__((ext_vector_type(8))) float v8f;
typedef int v4i __attribute__((vector_size(16)));
typedef __attribute__((address_space(1))) v4i gv4i;   // global 16B vector
typedef __attribute__((address_space(3))) v4i lv4i;   // LDS 16B vector
typedef __attribute__((address_space(1))) float gfloat;

__device__ __forceinline__ void atomAddF(float* p, float v) {
#if __has_builtin(__builtin_amdgcn_global_atomic_fadd_f32)
    __builtin_amdgcn_global_atomic_fadd_f32((gfloat*)p, v);
#else
    atomicAdd(p, v);
#endif
}

// ---------------------------------------------------------------------------
// 1) zero the accumulator (d_out) and init deg[i] = 1 (self-loop)
// ---------------------------------------------------------------------------
__global__ void __launch_bounds__(256) k_init(float* __restrict__ out,
                                              float* __restrict__ deg,
                                              int n_nodes) {
    long long tid = (long long)blockIdx.x * blockDim.x + threadIdx.x;
    long long total = (long long)n_nodes * DH;
    if (tid < total) out[tid] = 0.0f;
    if (tid < n_nodes) deg[(int)tid] = 1.0f;
}

// ---------------------------------------------------------------------------
// 2) degree count over edge targets (col)
// ---------------------------------------------------------------------------
__global__ void __launch_bounds__(256) k_count_deg(const int* __restrict__ col,
                                                   float* __restrict__ deg,
                                                   int n_edges) {
    int e = blockIdx.x * blockDim.x + threadIdx.x;
    if (e < n_edges) atomAddF(&deg[col[e]], 1.0f);
}

// ---------------------------------------------------------------------------
// 3) deg -> dinv in place
// ---------------------------------------------------------------------------
__global__ void __launch_bounds__(256) k_dinv(float* __restrict__ deg, int n_nodes) {
    int i = blockIdx.x * blockDim.x + threadIdx.x;
    if (i < n_nodes) {
        float d = deg[i];
        deg[i] = (d > 0.0f) ? rsqrtf(d) : 0.0f;
    }
}

// ---------------------------------------------------------------------------
// 4) h = x @ W  via fp32 WMMA 16x16x4, A staged in LDS with async copies.
//    Block = 256 threads = 8 waves; 16 output rows per block; wave w owns
//    column tile n0 = 16*w.  The 16x128 A tile is copied global->LDS once
//    (async b128, ASYNCcnt) instead of each wave re-loading it (8x traffic cut).
//    A lane layout: M = lane%16, K pair {koff, koff+1}, koff = 2*(lane/16).
//    B lane layout: N = n0 + lane%16, same K pair.
// ---------------------------------------------------------------------------
__global__ void __launch_bounds__(256, 2) k_gemm_wmma(const float* __restrict__ x,
                                                      const float* __restrict__ W,
                                                      float* __restrict__ h,
                                                      int n_nodes) {
    __shared__ float As[16 * A_STRIDE];

    const int lane = threadIdx.x & 31;
    const int wave = threadIdx.x >> 5;
    const int l16  = lane & 15;
    const int half = lane >> 4;          // 0 or 1
    const int koff = half * 2;           // K sub-offset for this half-wave
    const int row_base = blockIdx.x * 16;
    const int n0 = wave * 16;

    // ---- stage A tile: 16 rows x 128 floats = 512 chunks of 16B ----
#pragma unroll
    for (int i = 0; i < 2; ++i) {
        int chunk = threadIdx.x + i * 256;     // 0..511
        int r  = chunk >> 5;                   // row in tile
        int c4 = (chunk & 31) * 4;             // float offset in row
        int grow = row_base + r;
        if (grow >= n_nodes) grow = n_nodes - 1;
        const float* gsrc = x + (long long)grow * DH + c4;
        float* ldst = &As[r * A_STRIDE + c4];
#if __has_builtin(__builtin_amdgcn_global_load_async_to_lds_b128)
        __builtin_amdgcn_global_load_async_to_lds_b128(
            (gv4i*)const_cast<float*>(gsrc),
            (lv4i*)ldst,
            0, 0);
#else
        *(float4*)ldst = *(const float4*)gsrc;
#endif
    }
#if __has_builtin(__builtin_amdgcn_global_load_async_to_lds_b128)
#if __has_builtin(__builtin_amdgcn_s_wait_asynccnt)
    __builtin_amdgcn_s_wait_asynccnt(0);
#else
    asm volatile("s_wait_asynccnt 0" ::: "memory");
#endif
#endif
    __syncthreads();

    // ---- K loop: 32 x WMMA fp32 16x16x4, A from LDS, B from L2-resident W ----
    const float* __restrict__ arow = &As[l16 * A_STRIDE];
    v8f c = {};
#pragma unroll
    for (int kk = 0; kk < DH; kk += 4) {
        v2f a, b;
        a.x = arow[kk + koff];
        a.y = arow[kk + koff + 1];
        b.x = W[(kk + koff) * DH + n0 + l16];
        b.y = W[(kk + koff + 1) * DH + n0 + l16];
        c = __builtin_amdgcn_wmma_f32_16x16x4_f32(false, a, false, b,
                                                  (short)0, c, false, false);
    }

#pragma unroll
    for (int r = 0; r < 8; ++r) {
        int m = r + half * 8;            // C/D layout: VGPR r -> rows r / r+8
        int row = row_base + m;
        if (row < n_nodes)
            h[(long long)row * DH + n0 + l16] = c[r];
    }
}

// ---------------------------------------------------------------------------
// 5) edge scatter: one wave32 per edge, float4 per lane.
//    out[col] += h[row] * dinv[row]*dinv[col]   (native fp32 atomics, L2-resident)
// ---------------------------------------------------------------------------
__global__ void __launch_bounds__(256) k_scatter(const int* __restrict__ ei,
                                                 const float* __restrict__ h,
                                                 const float* __restrict__ dinv,
                                                 float* __restrict__ out,
                                                 int n_edges) {
    int e    = (blockIdx.x * blockDim.x + threadIdx.x) >> 5;
    int lane = threadIdx.x & 31;
    if (e >= n_edges) return;
    int row = ei[e];
    int col = ei[n_edges + e];
    float nrm = dinv[row] * dinv[col];

    float4 v = ((const float4*)(h + (long long)row * DH))[lane];
    float* ob = out + (long long)col * DH + lane * 4;
    atomAddF(ob + 0, v.x * nrm);
    atomAddF(ob + 1, v.y * nrm);
    atomAddF(ob + 2, v.z * nrm);
    atomAddF(ob + 3, v.w * nrm);
}

// ---------------------------------------------------------------------------
// 6) finalize: add self-loop term + bias, LayerNorm across 128, ReLU.
//    One wave32 per node; 4 channels per lane; shfl_xor butterfly reduction.
// ---------------------------------------------------------------------------
__global__ void __launch_bounds__(256) k_finalize(float* __restrict__ out,
                                                  const float* __restrict__ h,
                                                  const float* __restrict__ dinv,
                                                  const float* __restrict__ bias,
                                                  const float* __restrict__ gamma,
                                                  const float* __restrict__ beta,
                                                  int n_nodes) {
    int node = (blockIdx.x * blockDim.x + threadIdx.x) >> 5;
    int lane = threadIdx.x & 31;
    if (node >= n_nodes) return;

    float di = dinv[node];
    float sn = di * di;                          // self-loop norm

    float4 acc = ((float4*)(out + (long long)node * DH))[lane];
    float4 hv  = ((const float4*)(h + (long long)node * DH))[lane];
    float4 bv  = ((const float4*)bias)[lane];

    float4 v;
    v.x = acc.x + hv.x * sn + bv.x;
    v.y = acc.y + hv.y * sn + bv.y;
    v.z = acc.z + hv.z * sn + bv.z;
    v.w = acc.w + hv.w * sn + bv.w;

    float s1 = v.x + v.y + v.z + v.w;
    float s2 = v.x * v.x + v.y * v.y + v.z * v.z + v.w * v.w;
#pragma unroll
    for (int m = 16; m >= 1; m >>= 1) {
        s1 += __shfl_xor(s1, m, 32);
        s2 += __shfl_xor(s2, m, 32);
    }
    float mu   = s1 * (1.0f / DH);
    float var  = s2 * (1.0f / DH) - mu * mu;
    float rstd = rsqrtf(var + LN_EPS);

    float4 g  = ((const float4*)gamma)[lane];
    float4 be = ((const float4*)beta)[lane];
    v.x = fmaxf((v.x - mu) * rstd * g.x + be.x, 0.0f);
    v.y = fmaxf((v.y - mu) * rstd * g.y + be.y, 0.0f);
    v.z = fmaxf((v.z - mu) * rstd * g.z + be.z, 0.0f);
    v.w = fmaxf((v.w - mu) * rstd * g.w + be.w, 0.0f);

    ((float4*)(out + (long long)node * DH))[lane] = v;
}

// ---------------------------------------------------------------------------
extern "C" void kernel_launch(void* const* d_in, const int* in_sizes, int n_in,
                              void* d_out, int out_size, void* d_ws, size_t ws_size,
                              hipStream_t stream) {
    const float* x     = (const float*)d_in[0];
    const int*   ei    = (const int*)  d_in[1];
    const float* W     = (const float*)d_in[2];
    const float* bias  = (const float*)d_in[3];
    const float* gamma = (const float*)d_in[4];
    const float* beta  = (const float*)d_in[5];

    const int N = in_sizes[0] / DH;
    const int E = in_sizes[1] / 2;

    float* out = (float*)d_out;
    float* h   = (float*)d_ws;                     // N*DH floats
    float* deg = h + (size_t)N * DH;               // N floats (becomes dinv)

    long long total = (long long)N * DH;
    k_init<<<(int)((total + 255) / 256), 256, 0, stream>>>(out, deg, N);
    k_count_deg<<<(E + 255) / 256, 256, 0, stream>>>(ei + E, deg, E);
    k_dinv<<<(N + 255) / 256, 256, 0, stream>>>(deg, N);
    k_gemm_wmma<<<(N + 15) / 16, 256, 0, stream>>>(x, W, h, N);
    k_scatter<<<(E + 7) / 8, 256, 0, stream>>>(ei, h, deg, out, E);
    k_finalize<<<(N + 7) / 8, 256, 0, stream>>>(out, h, deg, bias, gamma, beta, N);
}